// LinearPoolNewsvendorLayer_26259430047930
// MI455X (gfx1250) — compile-verified
//
#include <hip/hip_runtime.h>
#include <math.h>

// Problem constants (match reference: M=8, B=2048, L=4096, Q=0.5)
#define M_IN    8
#define BATCH   2048
#define LSUP    4096
#define QF      0.5f
#define THREADS 256
#define SEG     (LSUP / THREADS)    // 16 elements/thread in scan & output phases

// Async double-buffer staging geometry
#define CHUNK   256                 // columns staged per chunk
#define NCHUNK  (LSUP / CHUNK)      // 16 chunks per batch row
#define ROWP    (CHUNK + 16)        // padded LDS row stride -> 16-bank shift per row

typedef float v2f __attribute__((ext_vector_type(2)));
typedef float v4f __attribute__((ext_vector_type(4)));
typedef float v8f __attribute__((ext_vector_type(8)));

// Async-to-LDS builtin wants pointers to 16-byte int vectors (per hipcc diag):
//   param 0: AS1 (global) i32x4*,  param 1: AS3 (LDS) i32x4*
typedef int i32x4 __attribute__((vector_size(16)));
typedef __attribute__((address_space(1))) i32x4 gi32x4;
typedef __attribute__((address_space(3))) i32x4 li32x4;

#if defined(__has_builtin)
#if __has_builtin(__builtin_amdgcn_global_load_async_to_lds_b128) && \
    __has_builtin(__builtin_amdgcn_s_wait_asynccnt)
#define USE_ASYNC_LDS 1
#endif
#endif
#ifndef USE_ASYNC_LDS
#define USE_ASYNC_LDS 0
#endif

// One workgroup per batch row b.
// Phase 1: pdf[b,:] = sum_m softmax(w)[m] * inputs[m,b,:] via chained
//          V_WMMA_F32_16X16X4_F32; inputs double-buffer-staged into LDS with
//          GLOBAL_LOAD_ASYNC_TO_LDS_B128 (ASYNCcnt pipeline) when available.
// Phase 2: segmented block scan of pdf row -> first CDF >= Q crossing -> z.
// Phase 3: vectorized emit of pdf / pinball / error (+ z scalar).
__global__ __launch_bounds__(THREADS) void pool_newsvendor_kernel(
    const float* __restrict__ inputs,   // [M, B, L]
    const float* __restrict__ support,  // [L]
    const float* __restrict__ weights,  // [M]
    float* __restrict__ out)            // pdf[B*L] | z[B] | pinball[B*L] | error[B*L]
{
    const int b    = blockIdx.x;
    const int t    = threadIdx.x;
    const int lane = t & 31;
    const int wv   = t >> 5;            // 8 waves per block

    __shared__ float s_pdf[LSUP];       // 16 KB pdf row
    __shared__ float s_scan[THREADS];
    __shared__ int   s_idx;
    __shared__ float s_z;
#if USE_ASYNC_LDS
    __shared__ float s_in[2][M_IN][ROWP];  // ~17 KB double-buffered input tile
#endif

    // ---- softmax(weights), uniform across all threads (8 scalar loads, cached)
    float w[M_IN];
    {
        float mx = weights[0];
        #pragma unroll
        for (int m = 1; m < M_IN; ++m) mx = fmaxf(mx, weights[m]);
        float sum = 0.f;
        #pragma unroll
        for (int m = 0; m < M_IN; ++m) { w[m] = __expf(weights[m] - mx); sum += w[m]; }
        const float inv = 1.f / sum;
        #pragma unroll
        for (int m = 0; m < M_IN; ++m) w[m] *= inv;
    }

    // WMMA A operand (16x4 f32, 2 VGPRs): lanes 0-15 hold K={0,1}, lanes 16-31
    // K={2,3}; all 16 rows carry the same weights -> all D rows identical.
    const int j = lane & 15;            // output column within a 16-wide tile
    const int h = lane >> 4;            // half-wave select (K pair)
    v2f a0, a1;
    a0.x = w[2 * h];     a0.y = w[2 * h + 1];      // contracts m = 0..3
    a1.x = w[4 + 2 * h]; a1.y = w[5 + 2 * h];      // contracts m = 4..7

    const size_t mstride = (size_t)BATCH * LSUP;   // stride between input PDFs
    const size_t rowoff  = (size_t)b * LSUP;

#if USE_ASYNC_LDS
    // ---- Phase 1 (async-staged): GLOBAL -> LDS b128 copies, WMMA from LDS.
    auto issue_chunk = [&](int ck, int buf) {
        // 8 rows x 256 cols = 512 float4s per chunk; 2 async b128 per thread
        // => 2 async instructions per wave per chunk (ASYNCcnt += 2).
        #pragma unroll
        for (int r = 0; r < 2; ++r) {
            const int idx = t + r * THREADS;
            const int m   = idx >> 6;           // CHUNK/4 = 64 float4 per row
            const int c4  = idx & 63;
            const float* src = inputs + (size_t)m * mstride + rowoff
                             + (size_t)ck * CHUNK + (size_t)c4 * 4;
            float* dst = &s_in[buf][m][c4 * 4];
            __builtin_amdgcn_global_load_async_to_lds_b128(
                (gi32x4*)src, (li32x4*)dst, 0, 0);
        }
    };

    issue_chunk(0, 0);
    for (int ck = 0; ck < NCHUNK; ++ck) {
        const int buf = ck & 1;
        if (ck + 1 < NCHUNK) {
            issue_chunk(ck + 1, buf ^ 1);
            __builtin_amdgcn_s_wait_asynccnt(2);   // chunk ck fully landed
        } else {
            __builtin_amdgcn_s_wait_asynccnt(0);
        }
        __syncthreads();                            // visible to all waves

        #pragma unroll
        for (int tt = 0; tt < CHUNK / (16 * 8); ++tt) {  // 2 tiles per wave
            const int tb = (wv * (CHUNK / (16 * 8)) + tt) * 16;
            // B operand (4x16 f32): lane -> (K = m, N = j); padded rows make
            // the half-wave row split conflict-free in LDS banks.
            v2f bb0, bb1;
            bb0.x = s_in[buf][2 * h][tb + j];
            bb0.y = s_in[buf][2 * h + 1][tb + j];
            bb1.x = s_in[buf][4 + 2 * h][tb + j];
            bb1.y = s_in[buf][5 + 2 * h][tb + j];
            v8f c = {};
            c = __builtin_amdgcn_wmma_f32_16x16x4_f32(false, a0, false, bb0,
                                                      (short)0, c, false, false);
            c = __builtin_amdgcn_wmma_f32_16x16x4_f32(false, a1, false, bb1,
                                                      (short)0, c, false, false);
            if (lane < 16) s_pdf[ck * CHUNK + tb + j] = c[0];
        }
        __syncthreads();   // done reading buf before it is overwritten
    }
#else
    // ---- Phase 1 (fallback): direct per-lane global gathers into B operand.
    #pragma unroll 1
    for (int it = 0; it < LSUP / (16 * 8); ++it) {
        const int    base = (it * 8 + wv) * 16;
        const size_t col  = rowoff + (size_t)(base + j);
        v2f bb0, bb1;
        bb0.x = inputs[(size_t)(2 * h)     * mstride + col];
        bb0.y = inputs[(size_t)(2 * h + 1) * mstride + col];
        bb1.x = inputs[(size_t)(4 + 2 * h) * mstride + col];
        bb1.y = inputs[(size_t)(5 + 2 * h) * mstride + col];
        __builtin_prefetch(&inputs[(size_t)(2 * h) * mstride + col + 128], 0, 1);
        v8f c = {};
        c = __builtin_amdgcn_wmma_f32_16x16x4_f32(false, a0, false, bb0,
                                                  (short)0, c, false, false);
        c = __builtin_amdgcn_wmma_f32_16x16x4_f32(false, a1, false, bb1,
                                                  (short)0, c, false, false);
        if (lane < 16) s_pdf[base + j] = c[0];
    }
#endif

    if (t == 0) s_idx = LSUP - 1;
    __syncthreads();

    // ---- Phase 2: segmented scan + first CDF >= Q crossing.
    const int segBase = t * SEG;
    float part = 0.f;
    #pragma unroll
    for (int i = 0; i < SEG; ++i) part += s_pdf[segBase + i];
    s_scan[t] = part;
    __syncthreads();
    #pragma unroll
    for (int off = 1; off < THREADS; off <<= 1) {
        float v = (t >= off) ? s_scan[t - off] : 0.f;
        __syncthreads();
        s_scan[t] += v;
        __syncthreads();
    }
    float run = s_scan[t] - part;       // exclusive prefix of this segment
    int cand = LSUP;
    #pragma unroll
    for (int i = 0; i < SEG; ++i) {
        run += s_pdf[segBase + i];
        if (run >= QF && cand == LSUP) cand = segBase + i;
    }
    if (cand < LSUP) atomicMin(&s_idx, cand);
    __syncthreads();

    if (t == 0) {
        float z = support[s_idx];
        z = fminf(fmaxf(z, 0.f), 1.f);
        s_z = z;
        out[(size_t)BATCH * LSUP + b] = z;   // z output slot
    }
    __syncthreads();
    const float z = s_z;

    // ---- Phase 3: emit pdf / pinball / error with 128-bit stores.
    float* __restrict__ pdf_out = out;
    float* __restrict__ pin_out = out + (size_t)BATCH * LSUP + BATCH;
    float* __restrict__ err_out = pin_out + (size_t)BATCH * LSUP;

    #pragma unroll
    for (int i = 0; i < SEG / 4; ++i) {
        const int l = segBase + i * 4;
        v4f sp = *(const v4f*)(support + l);
        v4f pd = *(const v4f*)(s_pdf + l);
        v4f e, p;
        #pragma unroll
        for (int k = 0; k < 4; ++k) {
            float ek = sp[k] - z;
            e[k] = ek;
            p[k] = fmaxf(QF * ek, (QF - 1.f) * ek);
        }
        *(v4f*)(pdf_out + rowoff + l) = pd;
        *(v4f*)(pin_out + rowoff + l) = p;
        *(v4f*)(err_out + rowoff + l) = e;
    }
}

extern "C" void kernel_launch(void* const* d_in, const int* in_sizes, int n_in,
                              void* d_out, int out_size, void* d_ws, size_t ws_size,
                              hipStream_t stream) {
    const float* inputs  = (const float*)d_in[0];  // [M,B,L] fp32
    const float* support = (const float*)d_in[1];  // [L] fp32
    const float* weights = (const float*)d_in[2];  // [M] fp32
    float* out = (float*)d_out;
    (void)in_sizes; (void)n_in; (void)out_size; (void)d_ws; (void)ws_size;

    pool_newsvendor_kernel<<<BATCH, THREADS, 0, stream>>>(inputs, support, weights, out);
}